// BeltramiPositionalEncoding_74423193305340
// MI455X (gfx1250) — compile-verified
//
#include <hip/hip_runtime.h>

// Beltrami diffusion positional encoding, MI455X (gfx1250).
//
// Roofline: 5-step, 5-point stencil over (4,8,1024,1024) fp32 = pure bandwidth.
// All 5 steps fused in-register (trapezoid tiling): HBM traffic ~330 MB
// (~14 us @ 23.3 TB/s) vs ~1.4 GB unfused. Register tiling (4x4 patch/thread,
// weights in VGPRs) cuts LDS to 8 b128 border-exchange ops per 16 points per
// step (0.5 ds-instr/point vs 10 in a naive LDS stencil), so LDS no longer
// dominates the HBM bound. Tile staging uses CDNA5 async global->LDS loads
// (ASYNCcnt / s_wait_asynccnt), overlapped with compute; outputs are
// nontemporal b128 stores.

#define ALPHA  2.0f
#define DT     0.1f
#define TSTEPS 5

#define OUT_T  56                 // exact output tile per block
#define CT     64                 // computed region (16x16 threads * 4x4)
#define LT     66                 // loaded region (computed + 1 halo)
#define LPITCH 68                 // LDS row pitch for tiles (floats)
#define NT     256                // 16x16 threads = 8 waves (wave32)
#define LPT    ((LT * LT + NT - 1) / NT)   // 18 async loads/thread/tile

static constexpr int Hs = 1024, Ws = 1024, Bs = 4, Ks = 8;
static constexpr int GB = (Hs + OUT_T - 1) / OUT_T;   // 19 blocks per dim

typedef float v4f __attribute__((ext_vector_type(4)));
typedef __attribute__((address_space(1))) int gi32;   // global int
typedef __attribute__((address_space(3))) int si32;   // LDS int

#if defined(__gfx1250__) && __has_builtin(__builtin_amdgcn_global_load_async_to_lds_b32)
  #define ASYNC_LD_B32(ldsp, gp)                                   \
    __builtin_amdgcn_global_load_async_to_lds_b32(                 \
        (gi32*)(gp), (si32*)(ldsp), 0, 0)
  #if __has_builtin(__builtin_amdgcn_s_wait_asynccnt)
    #define WAIT_ASYNC(n) __builtin_amdgcn_s_wait_asynccnt(n)
  #else
    #define WAIT_ASYNC(n) asm volatile("s_wait_asynccnt %0" :: "n"(n) : "memory")
  #endif
#else
  #define ASYNC_LD_B32(ldsp, gp) (*(ldsp) = *(gp))
  #define WAIT_ASYNC(n) ((void)0)
#endif

// Async-load a LT x LT tile (periodic wrap; H=W=1024 -> mask) into LDS.
// Clamped trip count: every wave issues exactly LPT loads -> exact ASYNCcnt
// bookkeeping (loads complete in order per wave).
__device__ __forceinline__ void load_tile_async(float* dst,
                                                const float* __restrict__ src,
                                                int oy0, int ox0, int tid) {
  #pragma unroll
  for (int j = 0; j < LPT; ++j) {
    int i = tid + j * NT;
    if (i >= LT * LT) i = LT * LT - 1;   // harmless dup, keeps count fixed
    int iy = i / LT;
    int ix = i - iy * LT;
    int gy = (oy0 + iy) & (Hs - 1);
    int gx = (ox0 + ix) & (Ws - 1);
    ASYNC_LD_B32(dst + iy * LPITCH + ix, src + (size_t)gy * Ws + gx);
  }
}

__global__ __launch_bounds__(NT)
void BeltramiPositionalEncoding_74423193305340_kernel(
    const float* __restrict__ lambda_edge,   // (B,1,H,W)
    const float* __restrict__ p_init,        // (B,K,H,W)
    float* __restrict__ p_out) {             // (B,K,H,W)
  __shared__ float sLe[LT * LPITCH];     // lambda tile (read once for weights)
  __shared__ float sTile[LT * LPITCH];   // per-channel p tile (async target)
  // Border-exchange buffers, double-buffered; one v4f per thread edge.
  // Column arrays padded to 17 so lane banks stride by 4 (conflict-free b128).
  __shared__ v4f sTop[2][16][16];        // [buf][ty][tx]: thread's top row
  __shared__ v4f sBot[2][16][16];        // [buf][ty][tx]: thread's bottom row
  __shared__ v4f sLft[2][16][17];        // [buf][tx][ty]: thread's left col
  __shared__ v4f sRgt[2][16][17];        // [buf][tx][ty]: thread's right col

  const int tid = threadIdx.x;
  const int tx  = tid & 15;
  const int ty  = tid >> 4;
  const int ox0 = (int)blockIdx.x * OUT_T - 5;   // global of region coord 0
  const int oy0 = (int)blockIdx.y * OUT_T - 5;
  const int b   = (int)blockIdx.z;

  // 1) Kick off async loads: lambda first, then p(k=0).
  load_tile_async(sLe, lambda_edge + (size_t)b * Hs * Ws, oy0, ox0, tid);
  load_tile_async(sTile, p_init + (size_t)b * Ks * Hs * Ws, oy0, ox0, tid);
  WAIT_ASYNC(LPT);    // first LPT (lambda) done; p(0) still in flight
  __syncthreads();

  // 2) Per-thread weights in registers, reused for all steps and channels.
  //    wh[j][xi] = w at horizontal edge (x=px+xi-1 .. px+xi) on row py+j
  //    wv[yi][i] = w at vertical   edge (y=py+yi-1 .. py+yi) on col px+i
  const int px = 1 + 4 * tx;    // patch base in region coords [1..61]
  const int py = 1 + 4 * ty;
  float wh[4][5], wv[5][4];
  #pragma unroll
  for (int j = 0; j < 4; ++j)
    #pragma unroll
    for (int xi = 0; xi < 5; ++xi) {
      float c = sLe[(py + j) * LPITCH + px + xi];
      float l = sLe[(py + j) * LPITCH + px + xi - 1];
      wh[j][xi] = 1.0f / (1.0f + ALPHA * fmaxf(c, l));
    }
  #pragma unroll
  for (int yi = 0; yi < 5; ++yi)
    #pragma unroll
    for (int i = 0; i < 4; ++i) {
      float c = sLe[(py + yi) * LPITCH + px + i];
      float u = sLe[(py + yi - 1) * LPITCH + px + i];
      wv[yi][i] = 1.0f / (1.0f + ALPHA * fmaxf(c, u));
    }

  float p[4][4];                 // 4x4 patch in registers
  float hT[4], hB[4], hL[4], hR[4];   // halo ring (edge threads keep stale)

  auto do_step = [&]() {
    float np[4][4];
    #pragma unroll
    for (int j = 0; j < 4; ++j)
      #pragma unroll
      for (int i = 0; i < 4; ++i) {
        float c  = p[j][i];
        float up = j     ? p[j - 1][i] : hT[i];
        float dn = j < 3 ? p[j + 1][i] : hB[i];
        float lf = i     ? p[j][i - 1] : hL[j];
        float rt = i < 3 ? p[j][i + 1] : hR[j];
        float dv = wh[j][i + 1] * (rt - c) - wh[j][i] * (c - lf)
                 + wv[j + 1][i] * (dn - c) - wv[j][i] * (c - up);
        np[j][i] = c + DT * dv;
      }
    #pragma unroll
    for (int j = 0; j < 4; ++j)
      #pragma unroll
      for (int i = 0; i < 4; ++i) p[j][i] = np[j][i];
  };

  // 3) Channels: 5 fused steps, borders exchanged via b128 LDS ops.
  for (int k = 0; k < Ks; ++k) {
    WAIT_ASYNC(0);     // p(k) tile resident in sTile
    __syncthreads();

    // Patch + initial halo from the tile (once per channel).
    #pragma unroll
    for (int j = 0; j < 4; ++j)
      #pragma unroll
      for (int i = 0; i < 4; ++i)
        p[j][i] = sTile[(py + j) * LPITCH + px + i];
    #pragma unroll
    for (int i = 0; i < 4; ++i) {
      hT[i] = sTile[(py - 1) * LPITCH + px + i];
      hB[i] = sTile[(py + 4) * LPITCH + px + i];
    }
    #pragma unroll
    for (int j = 0; j < 4; ++j) {
      hL[j] = sTile[(py + j) * LPITCH + px - 1];
      hR[j] = sTile[(py + j) * LPITCH + px + 4];
    }

    do_step();   // step 1

    for (int t = 1; t < TSTEPS; ++t) {
      int buf = t & 1;
      sTop[buf][ty][tx] = (v4f){p[0][0], p[0][1], p[0][2], p[0][3]};
      sBot[buf][ty][tx] = (v4f){p[3][0], p[3][1], p[3][2], p[3][3]};
      sLft[buf][tx][ty] = (v4f){p[0][0], p[1][0], p[2][0], p[3][0]};
      sRgt[buf][tx][ty] = (v4f){p[0][3], p[1][3], p[2][3], p[3][3]};
      __syncthreads();
      // All step-1 tile reads are consumed before this barrier -> safe to
      // overwrite sTile with the next channel's tile (hidden under steps 2-5).
      if (t == 1 && k + 1 < Ks)
        load_tile_async(sTile, p_init + ((size_t)b * Ks + (k + 1)) * Hs * Ws,
                        oy0, ox0, tid);
      if (ty > 0)  { v4f v = sBot[buf][ty - 1][tx];
                     hT[0]=v.x; hT[1]=v.y; hT[2]=v.z; hT[3]=v.w; }
      if (ty < 15) { v4f v = sTop[buf][ty + 1][tx];
                     hB[0]=v.x; hB[1]=v.y; hB[2]=v.z; hB[3]=v.w; }
      if (tx > 0)  { v4f v = sRgt[buf][tx - 1][ty];
                     hL[0]=v.x; hL[1]=v.y; hL[2]=v.z; hL[3]=v.w; }
      if (tx < 15) { v4f v = sLft[buf][tx + 1][ty];
                     hR[0]=v.x; hR[1]=v.y; hR[2]=v.z; hR[3]=v.w; }
      do_step();
    }

    // 4) Store the exact inner 56x56 (region coords [5..60]) -> only threads
    //    tx,ty in [1..14] hold valid points; nontemporal b128, guarded at the
    //    right/bottom partial blocks (output coverage is disjoint).
    if (tx >= 1 && tx <= 14 && ty >= 1 && ty <= 14) {
      float* outB = p_out + ((size_t)b * Ks + k) * Hs * Ws;
      int gx = ox0 + px;          // = bx*56 + 4*(tx-1), 16B-aligned
      if (gx < Ws) {
        #pragma unroll
        for (int j = 0; j < 4; ++j) {
          int gy = oy0 + py + j;  // = by*56 + 4*(ty-1) + j
          if (gy < Hs) {
            v4f v = (v4f){p[j][0], p[j][1], p[j][2], p[j][3]};
            __builtin_nontemporal_store(v, (v4f*)(outB + (size_t)gy * Ws + gx));
          }
        }
      }
    }
  }
}

extern "C" void kernel_launch(void* const* d_in, const int* in_sizes, int n_in,
                              void* d_out, int out_size, void* d_ws, size_t ws_size,
                              hipStream_t stream) {
  // setup_inputs order: x(0), lambda_edge(1), gxx(2), gxy(3), gyy(4), p_init(5)
  // x/gxx/gxy/gyy are unused by the reference.
  const float* lambda_edge = (const float*)d_in[1];
  const float* p_init      = (const float*)d_in[5];
  float*       out         = (float*)d_out;

  dim3 grid(GB, GB, Bs);   // 19 x 19 x 4
  dim3 block(NT);
  hipLaunchKernelGGL(BeltramiPositionalEncoding_74423193305340_kernel,
                     grid, block, 0, stream, lambda_edge, p_init, out);
}